// GraphSAGELayer_32298154066116
// MI455X (gfx1250) — compile-verified
//
#include <hip/hip_runtime.h>

typedef __attribute__((ext_vector_type(2))) float v2f;
typedef __attribute__((ext_vector_type(8))) float v8f;

#define N_NODES 100000
#define N_EDGES 1250000
#define DFEAT 64

// ---------------------------------------------------------------------------
// Zero the workspace (neigh accumulator + degree counts).
// ---------------------------------------------------------------------------
__global__ __launch_bounds__(256) void zero_ws_kernel(float* __restrict__ p, int n) {
    int i = blockIdx.x * 256 + threadIdx.x;
    if (i < n) p[i] = 0.0f;
}

// ---------------------------------------------------------------------------
// Scatter-add: one wave32 per edge. Lane l handles features 2l, 2l+1.
// x[src] row gathered as float2 (coalesced 256B per wave), scatter-added into
// neigh[dst] via L2-resident global_atomic_add_f32. Lane 0 bumps the degree.
// ---------------------------------------------------------------------------
__global__ __launch_bounds__(256) void scatter_edges_kernel(
    const float* __restrict__ x, const long long* __restrict__ ei,
    float* __restrict__ neigh, float* __restrict__ cnt) {
    int wave = (blockIdx.x * 256 + threadIdx.x) >> 5;
    int lane = threadIdx.x & 31;
    if (wave >= N_EDGES) return;
    long long src = ei[wave];            // row  (edge_index[0])
    long long dst = ei[N_EDGES + wave];  // col  (edge_index[1])
    const float* xs = x + (size_t)src * DFEAT;
    float* nd = neigh + (size_t)dst * DFEAT;
    v2f v = *(const v2f*)(xs + lane * 2);
    atomicAdd(nd + lane * 2 + 0, v.x);
    atomicAdd(nd + lane * 2 + 1, v.y);
    if (lane == 0) atomicAdd(cnt + dst, 1.0f);
}

// ---------------------------------------------------------------------------
// Mean normalization: neigh[i] /= max(count, 1).
// ---------------------------------------------------------------------------
__global__ __launch_bounds__(256) void normalize_kernel(
    float* __restrict__ neigh, const float* __restrict__ cnt) {
    int i = blockIdx.x * 256 + threadIdx.x;
    if (i >= N_NODES * DFEAT) return;
    float c = cnt[i >> 6];
    neigh[i] *= 1.0f / fmaxf(c, 1.0f);
}

// ---------------------------------------------------------------------------
// out = relu(concat(x, neigh) @ W + b) with V_WMMA_F32_16X16X4_F32.
// One wave per 16x16 output tile; K=128 consumed in 32 WMMA steps of K=4.
// K < 64 reads from x, K >= 64 reads from neigh (implicit concat).
//
// A (16x4 f32):  lane<16: {A[m][k0],A[m][k0+1]}, lane>=16: K offset +2.
// B (4x16 f32):  lane<16: {W[k0][n],W[k0+1][n]}, lane>=16: K offset +2.
// C/D (8 VGPRs): vgpr r -> M = row0 + r (+8 hi-half), N = col0 + lane%16.
// ---------------------------------------------------------------------------
__global__ __launch_bounds__(256) void sage_gemm_wmma_kernel(
    const float* __restrict__ x, const float* __restrict__ neigh,
    const float* __restrict__ W, const float* __restrict__ bias,
    float* __restrict__ out) {
    int tile = blockIdx.x * 8 + (threadIdx.x >> 5);
    int mt = tile >> 2;           // 6250 M-tiles
    int nt = tile & 3;            // 4 N-tiles
    int lane = threadIdx.x & 31;
    int l16 = lane & 15;
    bool hi = lane >= 16;
    int row0 = mt * 16;
    int col0 = nt * 16;
    int m = row0 + l16;           // A-fragment row for this lane
    int n = col0 + l16;           // B-fragment / output column for this lane

    const float* xr = x + (size_t)m * DFEAT;
    const float* nr = neigh + (size_t)m * DFEAT;

    v8f c = {};
#pragma unroll
    for (int ks = 0; ks < 32; ++ks) {
        int kk = ks * 4 + (hi ? 2 : 0);
        v2f a;
        if (ks < 16) a = *(const v2f*)(xr + kk);          // K in [0,64): self feats
        else         a = *(const v2f*)(nr + (kk - 64));   // K in [64,128): neighbor mean
        v2f b;
        b.x = W[(size_t)(kk + 0) * 64 + n];
        b.y = W[(size_t)(kk + 1) * 64 + n];
        c = __builtin_amdgcn_wmma_f32_16x16x4_f32(
            /*neg_a=*/false, a, /*neg_b=*/false, b,
            /*c_mod=*/(short)0, c, /*reuse_a=*/false, /*reuse_b=*/false);
    }

    float bv = bias[n];
#pragma unroll
    for (int r = 0; r < 8; ++r) {
        int mrow = row0 + r + (hi ? 8 : 0);
        float v = c[r] + bv;
        out[(size_t)mrow * 64 + n] = v > 0.0f ? v : 0.0f;
    }
}

// ---------------------------------------------------------------------------
// Launch
// ---------------------------------------------------------------------------
extern "C" void kernel_launch(void* const* d_in, const int* in_sizes, int n_in,
                              void* d_out, int out_size, void* d_ws, size_t ws_size,
                              hipStream_t stream) {
    const float*     x    = (const float*)d_in[0];
    const long long* ei   = (const long long*)d_in[1];  // int64 edge_index [2, E]
    const float*     W    = (const float*)d_in[2];      // [128, 64]
    const float*     bias = (const float*)d_in[3];      // [64]
    float* out = (float*)d_out;

    float* neigh = (float*)d_ws;                        // [N_NODES, 64]
    float* cnt   = neigh + (size_t)N_NODES * DFEAT;     // [N_NODES]

    // 1) zero accumulators (neigh + count)
    int nz = N_NODES * DFEAT + N_NODES;
    zero_ws_kernel<<<(nz + 255) / 256, 256, 0, stream>>>(neigh, nz);

    // 2) scatter-add neighbor features + degrees (8 edges per 256-thread block)
    scatter_edges_kernel<<<(N_EDGES + 7) / 8, 256, 0, stream>>>(x, ei, neigh, cnt);

    // 3) mean normalization
    normalize_kernel<<<(N_NODES * DFEAT + 255) / 256, 256, 0, stream>>>(neigh, cnt);

    // 4) fused concat + GEMM + bias + ReLU via f32 WMMA
    //    tiles = 6250 (M) * 4 (N) = 25000 waves; 8 waves/block -> 3125 blocks
    sage_gemm_wmma_kernel<<<3125, 256, 0, stream>>>(x, neigh, W, bias, out);
}